// RecurrentContinuousActionPolicy_33595234189692
// MI455X (gfx1250) — compile-verified
//
#include <hip/hip_runtime.h>
#include <math.h>

// Problem constants (reference: T,B,D,H,A = 512,256,256,256,8)
#define TT   512
#define BB   256
#define DD   256
#define HH   256
#define AA   8
#define NG   768   // 3*H
#define OUTC 16    // 2*A

typedef float v2f __attribute__((ext_vector_type(2)));
typedef float v8f __attribute__((ext_vector_type(8)));

#ifndef __has_builtin
#define __has_builtin(x) 0
#endif
#if __has_builtin(__builtin_amdgcn_tensor_load_to_lds) && \
    __has_builtin(__builtin_amdgcn_s_wait_tensorcnt)
#define USE_TDM 1
typedef unsigned int u32x4 __attribute__((ext_vector_type(4)));
typedef int          i32x4 __attribute__((ext_vector_type(4)));
typedef int          i32x8 __attribute__((ext_vector_type(8)));
#else
#define USE_TDM 0
#endif

static __device__ inline v8f vzero8() {
  v8f v;
#pragma unroll
  for (int i = 0; i < 8; ++i) v[i] = 0.0f;
  return v;
}

// =====================================================================
// Kernel 1: gates_x = x @ Wi + bi
//   M = T*B = 131072, K = D = 256, N = 3H = 768
//   256 threads = 8 wave32; WG tile 128x128; wave tile 32x64 (2x4 wmma tiles)
//   fp32 tensor path: v_wmma_f32_16x16x4_f32
//   B stored K-pair-interleaved in LDS so each fragment is ONE b64 load.
// =====================================================================
#define K1_KC   32
#define LDA_PAD 36    // A row stride (floats); 36*m mod 64 distinct for m<16
#define K1BP    144   // B pair-row stride (pairs); 144*8B mod 256B == 128B

__global__ __launch_bounds__(256)
void gates_gemm_kernel(const float* __restrict__ x,
                       const float* __restrict__ Wi,
                       const float* __restrict__ bi,
                       float* __restrict__ gx) {
  __shared__ float As[128][LDA_PAD];
  __shared__ float Bsp[K1_KC / 2][K1BP][2];   // [k/2][col][k&1]

  const int tid   = threadIdx.x;
  const int lane  = tid & 31;
  const int w     = tid >> 5;
  const int half  = lane >> 4;     // which 16-lane half
  const int lm    = lane & 15;
  const int waveM = w & 3;         // 4 wave rows  -> 4*32 = 128 M
  const int waveN = w >> 2;        // 2 wave cols  -> 2*64 = 128 N
  const long M0   = (long)blockIdx.x * 128;
  const int  N0   = blockIdx.y * 128;

  v8f acc[2][4];
#pragma unroll
  for (int i = 0; i < 2; ++i)
#pragma unroll
    for (int j = 0; j < 4; ++j) acc[i][j] = vzero8();

  for (int kc = 0; kc < DD; kc += K1_KC) {
    if (kc + K1_KC < DD) {  // hint next chunks into cache (global_prefetch_b8)
      __builtin_prefetch(x + (M0 + (tid >> 3)) * DD + kc + K1_KC, 0, 1);
      __builtin_prefetch(Wi + (long)(kc + K1_KC + (tid >> 5)) * NG + N0, 0, 1);
    }
    // stage A: 128 rows x 32 cols (x is [M, D] row-major, contiguous in K)
    {
      const int r0 = tid >> 3;         // 0..31
      const int c4 = (tid & 7) * 4;    // 0,4,...,28
#pragma unroll
      for (int p = 0; p < 4; ++p) {
        const int r = p * 32 + r0;
        const float4 va = *(const float4*)(x + (M0 + r) * DD + kc + c4);
        As[r][c4 + 0] = va.x; As[r][c4 + 1] = va.y;
        As[r][c4 + 2] = va.z; As[r][c4 + 3] = va.w;
      }
    }
    // stage B pair-interleaved: rows 2kp,2kp+1 x 4 cols per unit
    {
#pragma unroll
      for (int u = 0; u < 2; ++u) {
        const int uu = tid + u * 256;      // 512 units total
        const int kp = uu >> 5;            // 0..15 pair-row
        const int cg = (uu & 31) * 4;      // 0..124 col group
        const float4 va = *(const float4*)(Wi + (long)(kc + 2 * kp)     * NG + N0 + cg);
        const float4 vb = *(const float4*)(Wi + (long)(kc + 2 * kp + 1) * NG + N0 + cg);
        float* d = &Bsp[kp][cg][0];
        *(float4*)(d + 0) = make_float4(va.x, vb.x, va.y, vb.y);
        *(float4*)(d + 4) = make_float4(va.z, vb.z, va.w, vb.w);
      }
    }
    __syncthreads();

#pragma unroll
    for (int k0 = 0; k0 < K1_KC; k0 += 4) {
      v2f a[2], b[4];
#pragma unroll
      for (int i = 0; i < 2; ++i) {   // A frag: one aligned b64 LDS read
        const int r = waveM * 32 + i * 16 + lm;
        a[i] = *(const v2f*)&As[r][k0 + 2 * half];
      }
#pragma unroll
      for (int j = 0; j < 4; ++j) {   // B frag: one aligned b64 LDS read
        const int c = waveN * 64 + j * 16 + lm;
        b[j] = *(const v2f*)&Bsp[(k0 >> 1) + half][c][0];
      }
#pragma unroll
      for (int i = 0; i < 2; ++i)
#pragma unroll
        for (int j = 0; j < 4; ++j)
          acc[i][j] = __builtin_amdgcn_wmma_f32_16x16x4_f32(
              false, a[i], false, b[j], (short)0, acc[i][j], false, false);
    }
    __syncthreads();
  }

  // epilogue: C/D layout row = v + 8*half, col = lane&15
#pragma unroll
  for (int i = 0; i < 2; ++i) {
#pragma unroll
    for (int j = 0; j < 4; ++j) {
      const int c = N0 + waveN * 64 + j * 16 + lm;
      const float bias = bi[c];
#pragma unroll
      for (int v = 0; v < 8; ++v) {
        const long r = M0 + waveM * 32 + i * 16 + v + 8 * half;
        gx[r * NG + c] = acc[i][j][v] + bias;
      }
    }
  }
}

// =====================================================================
// Kernel 2: persistent GRU scan + fused head
//   16 WGs x 256 threads. WG (rb,cb) owns h rows rb*64..+63, cols cb*64..+63.
//   Wh slice stays K-pair-interleaved in LDS for all 512 steps.
//   Per-step h tile staged by the Tensor Data Mover (tensor_load_to_lds with
//   pad 4 dwords / 256 dwords -> LDS row stride 260), waited with
//   s_wait_tensorcnt; fp32 WMMA GEMM; GRU elementwise; ping-pong h;
//   atomic grid barrier; fused Dense(16)+clip/exp head -> d_out.
// =====================================================================
#define WHPP 208   // Wh pair-row stride (pairs); 208*8B mod 256B == 128B
#define HP   260   // h tile LDS row stride (floats) == 256 + 4 pad dwords
#define NWG  16

__global__ __launch_bounds__(256)
void gru_scan_kernel(const float* __restrict__ gx,
                     const float* __restrict__ Wh,
                     const float* __restrict__ bhn,
                     const float* __restrict__ Wo,
                     const float* __restrict__ bo,
                     float* __restrict__ hbuf,      // [2][B][H] ping-pong
                     unsigned* __restrict__ bar,    // bar[0]=ticket, bar[1]=gen
                     float* __restrict__ out) {     // [T][B][2A]
  extern __shared__ float smem[];
  float* WhSp = smem;                     // [128][WHPP][2] pair-interleaved
  float* hS   = smem + 128 * WHPP * 2;    // [64][HP]

  const int tid  = threadIdx.x;
  const int lane = tid & 31;
  const int w    = tid >> 5;
  const int half = lane >> 4;
  const int lm   = lane & 15;
  const int wm   = w >> 1;         // 4 wave rows  -> 4*16 = 64 M
  const int wn   = w & 1;          // 2 wave cols  -> 2*32 = 64 j per gate
  const int wg   = blockIdx.x;
  const int rb   = wg >> 2;        // batch-row block (64 rows)
  const int cb   = wg & 3;         // hidden-col block (64 cols)

  // Resident Wh slice, pair-interleaved: (k, g*256+cb*64+jl) -> WhSp[k/2][lc][k&1]
  for (int idx = tid; idx < 128 * 192; idx += 256) {
    const int kp = idx / 192;
    const int lc = idx % 192;
    const int g  = lc >> 6;
    const int jl = lc & 63;
    const long col = (long)g * HH + cb * 64 + jl;
    float* d = WhSp + ((long)kp * WHPP + lc) * 2;
    d[0] = Wh[(long)(2 * kp)     * NG + col];
    d[1] = Wh[(long)(2 * kp + 1) * NG + col];
  }

  for (int t = 0; t < TT; ++t) {
    const float* hsrc = hbuf + (size_t)(t & 1) * (BB * HH);
    float*       hdst = hbuf + (size_t)((t + 1) & 1) * (BB * HH);

    __syncthreads();  // previous-step readers done with hS
#if USE_TDM
    // --- TDM: DMA h rows [rb*64,+64) x 256 into hS with 4-dword row pad ---
    if (w == 0) {
      const unsigned long long ga =
          (unsigned long long)(const void*)(hsrc + (size_t)rb * 64 * HH);
      const unsigned ldsoff = (unsigned)(size_t)(void*)hS;  // flat low 32 = LDS offset
      u32x4 g0;
      g0[0] = 1u;                                   // count=1, user descriptor
      g0[1] = ldsoff;                               // lds_addr
      g0[2] = (unsigned)(ga & 0xFFFFFFFFull);       // global_addr[31:0]
      g0[3] = (unsigned)((ga >> 32) & 0x1FFFFFFull) // global_addr[56:32]
              | 0x80000000u;                        // type=2 ("image")
      i32x8 g1;
      g1[0] = (2 << 16)       // data_size = 4B
            | (1 << 20)       // pad_enable
            | (7 << 22)       // pad_interval: 256 dwords
            | (3 << 25);      // pad_amount: 4 dwords  -> row stride 260
      g1[1] = (HH & 0xFFFF) << 16;            // tensor_dim0[15:0]=256
      g1[2] = (HH & 0xFFFF) << 16;            // dim0 hi=0 | tensor_dim1[15:0]=256
      g1[3] = (HH & 0xFFFF) << 16;            // dim1 hi=0 | tile_dim0=256
      g1[4] = 64;                             // tile_dim1=64, tile_dim2=0
      g1[5] = HH;                             // tensor_dim0_stride = 256
      g1[6] = 0;
      g1[7] = 0;
      i32x4 gz4 = {0, 0, 0, 0};               // groups 2/3 unused (2D tensor)
      i32x8 gz8 = {0, 0, 0, 0, 0, 0, 0, 0};   // trailing group (clang-23 form)
      __builtin_amdgcn_tensor_load_to_lds(g0, g1, gz4, gz4, gz8, 0);
      __builtin_amdgcn_s_wait_tensorcnt(0);
    }
    __syncthreads();
#else
    for (int idx = tid; idx < 64 * 64; idx += 256) {
      const int r  = idx >> 6;
      const int c4 = (idx & 63) * 4;
      const float4 v = *(const float4*)(hsrc + (size_t)(rb * 64 + r) * HH + c4);
      float* d = hS + r * HP + c4;
      d[0] = v.x; d[1] = v.y; d[2] = v.z; d[3] = v.w;
    }
    __syncthreads();
#endif

    // gh = h @ Wh_slice : 6 accumulator tiles per wave (r0 r1 z0 z1 n0 n1)
    v8f acc[3][2];
#pragma unroll
    for (int g = 0; g < 3; ++g)
#pragma unroll
      for (int s = 0; s < 2; ++s) acc[g][s] = vzero8();

    const int ar = wm * 16 + lm;
#pragma unroll 2
    for (int k0 = 0; k0 < HH; k0 += 4) {
      const v2f a = *(const v2f*)&hS[ar * HP + k0 + 2 * half];
      const float* brow = WhSp + (long)((k0 >> 1) + half) * WHPP * 2;
#pragma unroll
      for (int g = 0; g < 3; ++g)
#pragma unroll
        for (int s = 0; s < 2; ++s) {
          const int c = g * 64 + wn * 32 + s * 16 + lm;
          const v2f b = *(const v2f*)(brow + c * 2);
          acc[g][s] = __builtin_amdgcn_wmma_f32_16x16x4_f32(
              false, a, false, b, (short)0, acc[g][s], false, false);
        }
    }

    // GRU elementwise: r,z,n live in matching accumulator slots
    const float* gxt = gx + (size_t)t * (BB * NG);
#pragma unroll
    for (int s = 0; s < 2; ++s) {
#pragma unroll
      for (int v = 0; v < 8; ++v) {
        const int lrow = wm * 16 + v + 8 * half;   // local row 0..63
        const int b_   = rb * 64 + lrow;
        const int j    = cb * 64 + wn * 32 + s * 16 + lm;
        const float* gxrow = gxt + (size_t)b_ * NG;
        const float gxr = gxrow[j];
        const float gxz = gxrow[j + HH];
        const float gxn = gxrow[j + 2 * HH];
        const float r = 1.0f / (1.0f + expf(-(gxr + acc[0][s][v])));
        const float z = 1.0f / (1.0f + expf(-(gxz + acc[1][s][v])));
        const float n = tanhf(gxn + r * (acc[2][s][v] + bhn[j]));
        const float hp = hS[lrow * HP + j];
        hdst[(size_t)b_ * HH + j] = (1.0f - z) * n + z * hp;
      }
    }

    // ----- grid-wide barrier (monotonic ticket; 16 co-resident WGs) -----
    __threadfence();
    __syncthreads();
    if (tid == 0) {
      const unsigned want = (unsigned)(t + 1) * NWG;
      const unsigned ticket = atomicAdd(&bar[0], 1u) + 1u;
      if (ticket == want) atomicExch(&bar[1], (unsigned)(t + 1));
      while (atomicAdd(&bar[1], 0u) < (unsigned)(t + 1)) {
        __builtin_amdgcn_s_sleep(2);
      }
    }
    __syncthreads();
    __threadfence();

    // ----- fused head for this step: rows wg*16..+15, one output/thread ---
    {
      const int row = wg * 16 + (tid >> 4);
      const int col = tid & 15;
      const float* hr = hdst + (size_t)row * HH;
      float sum = bo[col];
#pragma unroll 8
      for (int kk = 0; kk < HH; ++kk)
        sum = fmaf(hr[kk], Wo[kk * OUTC + col], sum);
      const float val = (col < AA)
                            ? sum
                            : expf(fminf(fmaxf(sum, -20.0f), 2.0f));
      out[((size_t)t * BB + row) * OUTC + col] = val;
    }
  }
}

// =====================================================================
// Host launch
// =====================================================================
extern "C" void kernel_launch(void* const* d_in, const int* in_sizes, int n_in,
                              void* d_out, int out_size, void* d_ws, size_t ws_size,
                              hipStream_t stream) {
  const float* x      = (const float*)d_in[0];
  const float* carry0 = (const float*)d_in[1];
  const float* Wi     = (const float*)d_in[2];
  const float* bi     = (const float*)d_in[3];
  const float* Wh     = (const float*)d_in[4];
  const float* bhn    = (const float*)d_in[5];
  const float* Wo     = (const float*)d_in[6];
  const float* bo     = (const float*)d_in[7];
  float* out = (float*)d_out;

  // workspace layout: [0,256) barrier state | ping-pong h | gates_x
  char*     ws   = (char*)d_ws;
  unsigned* bar  = (unsigned*)ws;
  float*    hbuf = (float*)(ws + 256);
  float*    gxb  = (float*)(ws + 256 + (size_t)2 * BB * HH * sizeof(float));

  (void)hipMemsetAsync(bar, 0, 256, stream);
  (void)hipMemcpyAsync(hbuf, carry0, (size_t)BB * HH * sizeof(float),
                       hipMemcpyDeviceToDevice, stream);

  // Kernel 1: gates_x = x @ Wi + bi
  dim3 g1((TT * BB) / 128, NG / 128);  // 1024 x 6
  gates_gemm_kernel<<<g1, 256, 0, stream>>>(x, Wi, bi, gxb);

  // Kernel 2: persistent scan (Wh pair slice + padded h tile resident in LDS)
  const size_t lds = (size_t)(128 * WHPP * 2 + 64 * HP) * sizeof(float);
  (void)hipFuncSetAttribute((const void*)gru_scan_kernel,
                            hipFuncAttributeMaxDynamicSharedMemorySize,
                            (int)lds);
  gru_scan_kernel<<<NWG, 256, lds, stream>>>(gxb, Wh, bhn, Wo, bo, hbuf, bar, out);
}